// GAT_42288247996702
// MI455X (gfx1250) — compile-verified
//
#include <hip/hip_runtime.h>
#include <hip/hip_bf16.h>
#include <math.h>

typedef __attribute__((ext_vector_type(2))) float v2f;
typedef __attribute__((ext_vector_type(8))) float v8f;
typedef __attribute__((ext_vector_type(4))) unsigned u32x4;
typedef __attribute__((ext_vector_type(8))) int i32x8;
typedef __attribute__((ext_vector_type(4))) int i32x4;

#define NN 50000
#define EE 800000
#define GG 128
#define ET (EE + NN)   /* edges incl. self loops */

#define HAVE_TDM __has_builtin(__builtin_amdgcn_tensor_load_to_lds)

// ---------- helpers ----------
__device__ __forceinline__ unsigned f2ord(float f) {
  unsigned u = __float_as_uint(f);
  return (u & 0x80000000u) ? ~u : (u | 0x80000000u);
}
__device__ __forceinline__ float ord2f(unsigned u) {
  return __uint_as_float((u & 0x80000000u) ? (u ^ 0x80000000u) : ~u);
}
__device__ __forceinline__ float lrelu(float x) { return x > 0.f ? x : 0.2f * x; }
__device__ __forceinline__ float elu1(float x)  { return x > 0.f ? x : (expf(x) - 1.f); }

#if HAVE_TDM
// 1-D contiguous TDM copy: nElems fp32 from global -> LDS.
// D# layout per CDNA5 ISA (group0/group1; groups 2,3 zero => <=2D tensor).
// 6-arg builtin form (clang-23 / therock-10.0 lane).
__device__ __forceinline__ void tdm_load_1d(unsigned lds_off, const void* gptr, unsigned nElems) {
  unsigned long long ga = (unsigned long long)(uintptr_t)gptr;
  u32x4 g0;
  g0.x = 1u;                                                  // count=1, no gather
  g0.y = lds_off;                                             // lds_addr (bytes)
  g0.z = (unsigned)ga;                                        // global_addr[31:0]
  g0.w = ((unsigned)(ga >> 32) & 0x01FFFFFFu) | (2u << 30);   // global_addr[56:32] | type=2
  i32x8 g1;
  g1[0] = (int)(2u << 16);                                    // wg_mask=0, data_size=2 (4B)
  g1[1] = (int)((nElems & 0xFFFFu) << 16);                    // tensor_dim0[15:0]
  g1[2] = (int)(nElems >> 16);                                // tensor_dim0[31:16], tensor_dim1=0
  g1[3] = (int)((nElems & 0xFFFFu) << 16);                    // tile_dim0
  g1[4] = 0;                                                  // tile_dim1=0 (1D), tile_dim2=0
  g1[5] = (int)nElems;                                        // tensor_dim0_stride[31:0]
  g1[6] = 0;
  g1[7] = 0;
  i32x4 z4 = {0, 0, 0, 0};
  i32x8 z8 = {0, 0, 0, 0, 0, 0, 0, 0};
  __builtin_amdgcn_tensor_load_to_lds(g0, g1, z4, z4, z8, 0);
}
#endif

// ---------- zero scratch accumulators ----------
__global__ void k_zero(unsigned* __restrict__ p, size_t n) {
  size_t i = (size_t)blockIdx.x * blockDim.x + threadIdx.x;
  size_t stride = (size_t)gridDim.x * blockDim.x;
  for (; i < n; i += stride) p[i] = 0u;
}

// ---------- in-degree + edge_attr sum per dst ----------
__global__ __launch_bounds__(256) void k_edge_deg(const int* __restrict__ ei,
                                                  const float* __restrict__ ea,
                                                  float* __restrict__ cnt,
                                                  float* __restrict__ sumea) {
  int e = blockIdx.x * 256 + threadIdx.x;
  if (e >= EE) return;
  int d = ei[EE + e];
  atomicAdd(cnt + d, 1.f);
  atomicAdd(sumea + d, ea[e]);
}

__global__ __launch_bounds__(256) void k_loop_mean(float* __restrict__ sumea,
                                                   const float* __restrict__ cnt) {
  int n = blockIdx.x * 256 + threadIdx.x;
  if (n >= NN) return;
  sumea[n] = sumea[n] / fmaxf(cnt[n], 1.f);   // becomes loop_ea
}

// ---------- layer1 linears: x[N,4] @ W[4,128] via one fp32 WMMA per tile ----------
__global__ __launch_bounds__(256) void k_gemm1(const float* __restrict__ x,
                                               const float* __restrict__ Wl,
                                               const float* __restrict__ bl,
                                               const float* __restrict__ Wr,
                                               const float* __restrict__ br,
                                               float* __restrict__ xl1,
                                               float* __restrict__ xr1) {
  int tid  = threadIdx.x;
  int wave = tid >> 5;
  int lane = tid & 31;
  int half = lane >> 4;        // 0: K=0,1   1: K=2,3
  int l    = lane & 15;
  int row  = blockIdx.x * 16 + l;           // A: M = l   (N==50000 % 16 == 0)
  int col  = wave * 16 + l;                 // B/D column

  v2f a  = { x[row * 4 + 2 * half], x[row * 4 + 2 * half + 1] };
  v2f bL = { Wl[(2 * half) * 128 + col], Wl[(2 * half + 1) * 128 + col] };
  v2f bR = { Wr[(2 * half) * 128 + col], Wr[(2 * half + 1) * 128 + col] };
  v8f cz = { 0.f, 0.f, 0.f, 0.f, 0.f, 0.f, 0.f, 0.f };

  v8f dL = __builtin_amdgcn_wmma_f32_16x16x4_f32(false, a, false, bL, (short)0, cz, false, false);
  v8f dR = __builtin_amdgcn_wmma_f32_16x16x4_f32(false, a, false, bR, (short)0, cz, false, false);

  float biasL = bl[col], biasR = br[col];
  int mbase = blockIdx.x * 16 + half * 8;
#pragma unroll
  for (int v = 0; v < 8; ++v) {
    int r = mbase + v;
    xl1[(size_t)r * 128 + col] = dL[v] + biasL;
    xr1[(size_t)r * 128 + col] = dR[v] + biasR;
  }
}

// ---------- layer1 edge pass 1: logits + segment max ----------
__global__ __launch_bounds__(256) void k_edge1_logits(const int* __restrict__ ei,
                                                      const float* __restrict__ ea,
                                                      const float* __restrict__ loopea,
                                                      const float* __restrict__ xl1,
                                                      const float* __restrict__ xr1,
                                                      const float* __restrict__ We1,
                                                      const float* __restrict__ att1,
                                                      float* __restrict__ lg1,
                                                      unsigned* __restrict__ mx1) {
  int t = blockIdx.x * 256 + threadIdx.x;     // (edge, head)
  if (t >= ET * 8) return;
  int e = t >> 3, h = t & 7;
  int s, d; float av;
  if (e < EE) { s = ei[e]; d = ei[EE + e]; av = ea[e]; }
  else        { s = d = e - EE;            av = loopea[s]; }
  const float* xs = xl1 + (size_t)s * 128 + h * 16;
  const float* xd = xr1 + (size_t)d * 128 + h * 16;
  const float* we = We1 + h * 16;
  const float* at = att1 + h * 16;
  float logit = 0.f;
#pragma unroll
  for (int c = 0; c < 16; ++c) {
    float m = lrelu(xs[c] + xd[c] + av * we[c]);
    logit += m * at[c];
  }
  lg1[t] = logit;
  atomicMax(mx1 + (size_t)d * 8 + h, f2ord(logit));
}

// ---------- layer1 edge pass 2: exp + segment denom ----------
__global__ __launch_bounds__(256) void k_edge1_norm(const int* __restrict__ ei,
                                                    float* __restrict__ lg1,
                                                    const unsigned* __restrict__ mx1,
                                                    float* __restrict__ den1) {
  int t = blockIdx.x * 256 + threadIdx.x;
  if (t >= ET * 8) return;
  int e = t >> 3, h = t & 7;
  int d = (e < EE) ? ei[EE + e] : (e - EE);
  float a = expf(lg1[t] - ord2f(mx1[(size_t)d * 8 + h]));
  lg1[t] = a;                                  // overwrite with unnormalized weight
  atomicAdd(den1 + (size_t)d * 8 + h, a);
}

// ---------- layer1 edge pass 3: weighted scatter-add of lin_l(x_src) ----------
__global__ __launch_bounds__(256) void k_edge1_agg(const int* __restrict__ ei,
                                                   const float* __restrict__ lg1,
                                                   const float* __restrict__ den1,
                                                   const float* __restrict__ xl1,
                                                   float* __restrict__ agg1) {
  int t = blockIdx.x * 256 + threadIdx.x;
  if (t >= ET * 8) return;
  int e = t >> 3, h = t & 7;
  int s, d;
  if (e < EE) { s = ei[e]; d = ei[EE + e]; }
  else        { s = d = e - EE; }
  float alpha = lg1[t] / den1[(size_t)d * 8 + h];
  const float* xs = xl1 + (size_t)s * 128 + h * 16;
  float* dst = agg1 + (size_t)d * 128 + h * 16;
  __builtin_prefetch(xs, 0, 0);
#pragma unroll
  for (int c = 0; c < 16; ++c) atomicAdd(dst + c, alpha * xs[c]);
}

// ---------- h1 = elu(agg1 + bias1) ----------
__global__ __launch_bounds__(256) void k_h1_elu(const float* __restrict__ agg1,
                                                const float* __restrict__ bias1,
                                                float* __restrict__ h1) {
  int t = blockIdx.x * 256 + threadIdx.x;
  if (t >= NN * 128) return;
  h1[t] = elu1(agg1[t] + bias1[t & 127]);
}

// ---------- layer2 linears: h1[N,128] @ [Wl2|Wr2][128,16] via WMMA K-loop ----------
// Weights staged into LDS by the Tensor Data Mover (TDM) when available.
__global__ __launch_bounds__(256) void k_gemm2(const float* __restrict__ h1,
                                               const float* __restrict__ Wl2,
                                               const float* __restrict__ bl2,
                                               const float* __restrict__ Wr2,
                                               const float* __restrict__ br2,
                                               float* __restrict__ xl2,
                                               float* __restrict__ xr2) {
  __shared__ float BsL[128 * 8];   // Wl2, row-major [k][c] (byte-identical to global)
  __shared__ float BsR[128 * 8];   // Wr2, row-major [k][c]
  int tid = threadIdx.x;

#if HAVE_TDM
  if (tid < 32) {                  // TDM ignores EXEC; issue once from wave 0
    tdm_load_1d((unsigned)(uintptr_t)(&BsL[0]), Wl2, 128 * 8);
    tdm_load_1d((unsigned)(uintptr_t)(&BsR[0]), Wr2, 128 * 8);
    __builtin_amdgcn_s_wait_tensorcnt(0);
  }
#else
  for (int i = tid; i < 128 * 8; i += 256) { BsL[i] = Wl2[i]; BsR[i] = Wr2[i]; }
#endif
  __syncthreads();

  int wave = tid >> 5;
  int lane = tid & 31;
  int half = lane >> 4;
  int l    = lane & 15;
  int rowTile = blockIdx.x * 128 + wave * 16;
  int row  = rowTile + l;
  int rowc = row < NN ? row : NN - 1;     // clamp: junk rows only touch unstored D rows
  const float* hrow = h1 + (size_t)rowc * 128;
  const float* bcol = (l < 8) ? (BsL + l) : (BsR + (l - 8));

  v8f acc = { 0.f, 0.f, 0.f, 0.f, 0.f, 0.f, 0.f, 0.f };
#pragma unroll 8
  for (int kk = 0; kk < 32; ++kk) {
    int k0 = kk * 4 + 2 * half;
    v2f a = { hrow[k0], hrow[k0 + 1] };
    v2f b = { bcol[k0 * 8], bcol[(k0 + 1) * 8] };
    acc = __builtin_amdgcn_wmma_f32_16x16x4_f32(false, a, false, b, (short)0, acc, false, false);
  }

#pragma unroll
  for (int v = 0; v < 8; ++v) {
    int r = rowTile + half * 8 + v;
    if (r < NN) {
      if (l < 8) xl2[(size_t)r * 8 + l]       = acc[v] + bl2[l];
      else       xr2[(size_t)r * 8 + (l - 8)] = acc[v] + br2[l - 8];
    }
  }
}

// ---------- layer2 edge pass 1 ----------
__global__ __launch_bounds__(256) void k_edge2_logits(const int* __restrict__ ei,
                                                      const float* __restrict__ ea,
                                                      const float* __restrict__ loopea,
                                                      const float* __restrict__ xl2,
                                                      const float* __restrict__ xr2,
                                                      const float* __restrict__ We2,
                                                      const float* __restrict__ att2,
                                                      float* __restrict__ lg2,
                                                      unsigned* __restrict__ mx2) {
  int e = blockIdx.x * 256 + threadIdx.x;
  if (e >= ET) return;
  int s, d; float av;
  if (e < EE) { s = ei[e]; d = ei[EE + e]; av = ea[e]; }
  else        { s = d = e - EE;            av = loopea[s]; }
  const float* xs = xl2 + (size_t)s * 8;
  const float* xd = xr2 + (size_t)d * 8;
  float logit = 0.f;
#pragma unroll
  for (int c = 0; c < 8; ++c) {
    float m = lrelu(xs[c] + xd[c] + av * We2[c]);
    logit += m * att2[c];
  }
  lg2[e] = logit;
  atomicMax(mx2 + d, f2ord(logit));
}

// ---------- layer2 edge pass 2 ----------
__global__ __launch_bounds__(256) void k_edge2_norm(const int* __restrict__ ei,
                                                    float* __restrict__ lg2,
                                                    const unsigned* __restrict__ mx2,
                                                    float* __restrict__ den2) {
  int e = blockIdx.x * 256 + threadIdx.x;
  if (e >= ET) return;
  int d = (e < EE) ? ei[EE + e] : (e - EE);
  float a = expf(lg2[e] - ord2f(mx2[d]));
  lg2[e] = a;
  atomicAdd(den2 + d, a);
}

// ---------- layer2 edge pass 3 ----------
__global__ __launch_bounds__(256) void k_edge2_agg(const int* __restrict__ ei,
                                                   const float* __restrict__ lg2,
                                                   const float* __restrict__ den2,
                                                   const float* __restrict__ xl2,
                                                   float* __restrict__ agg2) {
  int e = blockIdx.x * 256 + threadIdx.x;
  if (e >= ET) return;
  int s, d;
  if (e < EE) { s = ei[e]; d = ei[EE + e]; }
  else        { s = d = e - EE; }
  float alpha = lg2[e] / den2[d];
  const float* xs = xl2 + (size_t)s * 8;
  float* dst = agg2 + (size_t)d * 8;
#pragma unroll
  for (int c = 0; c < 8; ++c) atomicAdd(dst + c, alpha * xs[c]);
}

// ---------- h2 = elu(agg2 + bias2); mean-pool accumulate per graph ----------
__global__ __launch_bounds__(256) void k_pool(const float* __restrict__ agg2,
                                              const float* __restrict__ bias2,
                                              const int* __restrict__ batch,
                                              float* __restrict__ pooled,
                                              float* __restrict__ gcnt) {
  int n = blockIdx.x * 256 + threadIdx.x;
  if (n >= NN) return;
  int g = batch[n];
  atomicAdd(gcnt + g, 1.f);
#pragma unroll
  for (int c = 0; c < 8; ++c)
    atomicAdd(pooled + (size_t)g * 8 + c, elu1(agg2[(size_t)n * 8 + c] + bias2[c]));
}

// ---------- final: (pooled/cnt) @ W3 + b3 ----------
__global__ void k_final(const float* __restrict__ pooled,
                        const float* __restrict__ gcnt,
                        const float* __restrict__ W3,
                        const float* __restrict__ b3,
                        float* __restrict__ out) {
  int g = threadIdx.x;
  if (g >= GG) return;
  float inv = 1.f / fmaxf(gcnt[g], 1.f);
  float acc = b3[0];
#pragma unroll
  for (int c = 0; c < 8; ++c) acc += pooled[g * 8 + c] * inv * W3[c];
  out[g] = acc;
}

extern "C" void kernel_launch(void* const* d_in, const int* in_sizes, int n_in,
                              void* d_out, int out_size, void* d_ws, size_t ws_size,
                              hipStream_t stream) {
  const float* x     = (const float*)d_in[0];
  const int*   ei    = (const int*)  d_in[1];
  const float* ea    = (const float*)d_in[2];
  const int*   batch = (const int*)  d_in[3];
  const float* Wl1 = (const float*)d_in[4];  const float* bl1 = (const float*)d_in[5];
  const float* Wr1 = (const float*)d_in[6];  const float* br1 = (const float*)d_in[7];
  const float* We1 = (const float*)d_in[8];  const float* att1= (const float*)d_in[9];
  const float* bias1 = (const float*)d_in[10];
  const float* Wl2 = (const float*)d_in[11]; const float* bl2 = (const float*)d_in[12];
  const float* Wr2 = (const float*)d_in[13]; const float* br2 = (const float*)d_in[14];
  const float* We2 = (const float*)d_in[15]; const float* att2= (const float*)d_in[16];
  const float* bias2 = (const float*)d_in[17];
  const float* W3  = (const float*)d_in[18]; const float* b3  = (const float*)d_in[19];
  float* out = (float*)d_out;

  float* ws = (float*)d_ws;
  size_t off = 0;
  float*    cnt   = ws + off;              off += NN;
  float*    sumea = ws + off;              off += NN;              // -> loop_ea
  unsigned* mx1   = (unsigned*)(ws + off); off += (size_t)NN * 8;
  float*    den1  = ws + off;              off += (size_t)NN * 8;
  float*    agg1  = ws + off;              off += (size_t)NN * 128;
  unsigned* mx2   = (unsigned*)(ws + off); off += NN;
  float*    den2  = ws + off;              off += NN;
  float*    agg2  = ws + off;              off += (size_t)NN * 8;
  float*    pooled= ws + off;              off += (size_t)GG * 8;
  float*    gcnt  = ws + off;              off += GG;
  size_t nzero = off;                                              // words to zero
  float*    xl1   = ws + off;              off += (size_t)NN * 128;
  float*    xr1   = ws + off;              off += (size_t)NN * 128;
  float*    lg1   = ws + off;              off += (size_t)ET * 8;
  float*    h1    = ws + off;              off += (size_t)NN * 128;
  float*    xl2   = ws + off;              off += (size_t)NN * 8;
  float*    xr2   = ws + off;              off += (size_t)NN * 8;
  float*    lg2   = ws + off;              off += (size_t)ET;

  k_zero<<<2048, 256, 0, stream>>>((unsigned*)ws, nzero);
  k_edge_deg<<<(EE + 255) / 256, 256, 0, stream>>>(ei, ea, cnt, sumea);
  k_loop_mean<<<(NN + 255) / 256, 256, 0, stream>>>(sumea, cnt);
  k_gemm1<<<NN / 16, 256, 0, stream>>>(x, Wl1, bl1, Wr1, br1, xl1, xr1);
  k_edge1_logits<<<((size_t)ET * 8 + 255) / 256, 256, 0, stream>>>(ei, ea, sumea, xl1, xr1, We1, att1, lg1, mx1);
  k_edge1_norm<<<((size_t)ET * 8 + 255) / 256, 256, 0, stream>>>(ei, lg1, mx1, den1);
  k_edge1_agg<<<((size_t)ET * 8 + 255) / 256, 256, 0, stream>>>(ei, lg1, den1, xl1, agg1);
  k_h1_elu<<<((size_t)NN * 128 + 255) / 256, 256, 0, stream>>>(agg1, bias1, h1);
  k_gemm2<<<(NN + 127) / 128, 256, 0, stream>>>(h1, Wl2, bl2, Wr2, br2, xl2, xr2);
  k_edge2_logits<<<(ET + 255) / 256, 256, 0, stream>>>(ei, ea, sumea, xl2, xr2, We2, att2, lg2, mx2);
  k_edge2_norm<<<(ET + 255) / 256, 256, 0, stream>>>(ei, lg2, mx2, den2);
  k_edge2_agg<<<(ET + 255) / 256, 256, 0, stream>>>(ei, lg2, den2, xl2, agg2);
  k_pool<<<(NN + 255) / 256, 256, 0, stream>>>(agg2, bias2, batch, pooled, gcnt);
  k_final<<<1, 128, 0, stream>>>(pooled, gcnt, W3, b3, out);
}